// GCN_14113262535098
// MI455X (gfx1250) — compile-verified
//
#include <hip/hip_runtime.h>
#include <cstdint>

typedef __attribute__((ext_vector_type(2))) float v2f;
typedef __attribute__((ext_vector_type(8))) float v8f;

#define N_NODES 50000
#define F_IN 256
#define HID 128
#define CLS 32

// ---------------- degree / normalization ----------------
__global__ __launch_bounds__(256) void k_zero(float* __restrict__ p, int n) {
  int i = blockIdx.x * 256 + threadIdx.x;
  if (i < n) p[i] = 0.0f;
}

__global__ __launch_bounds__(256) void k_count(const int* __restrict__ dst,
                                               float* __restrict__ deg, int E) {
  int e = blockIdx.x * 256 + threadIdx.x;
  if (e < E) unsafeAtomicAdd(&deg[dst[e]], 1.0f);
}

__global__ __launch_bounds__(256) void k_rsqrt(float* __restrict__ d, int n) {
  int i = blockIdx.x * 256 + threadIdx.x;
  if (i < n) d[i] = rsqrtf(d[i] + 1.0f);  // +1 = self loop
}

// ---------------- WMMA f32 GEMM: H[N,M] = X[N,K] @ W[K,M] ----------------
// One wave computes a 16-row x M-col strip using M/16 accumulators.
// Layouts per CDNA5 ISA 7.12.2 (f32):
//   A (16x4):  lane L -> row L%16, vgpr v -> K = 2*(L/16) + v
//   B (4x16):  lane L -> col L%16, vgpr v -> K = 2*(L/16) + v
//   C/D (16x16): lane L -> col L%16, vgpr r -> row r + 8*(L/16)
template <int K, int M>
__global__ __launch_bounds__(128) void k_gemm(const float* __restrict__ X,
                                              const float* __restrict__ W,
                                              float* __restrict__ H,
                                              int nstrips) {
  constexpr int MT = M / 16;
  int strip = blockIdx.x * 4 + (threadIdx.x >> 5);
  if (strip >= nstrips) return;
  int lane = threadIdx.x & 31;
  int half = lane >> 4;
  int l16  = lane & 15;
  int n0   = strip * 16;

  v8f acc[MT];
  v8f zero = {0.f, 0.f, 0.f, 0.f, 0.f, 0.f, 0.f, 0.f};
#pragma unroll
  for (int m = 0; m < MT; ++m) acc[m] = zero;

  const float* xrow = X + (size_t)(n0 + l16) * K + 2 * half;
  for (int k0 = 0; k0 < K; k0 += 4) {
    v2f a = *(const v2f*)(xrow + k0);
    const float* wk = W + (size_t)(k0 + 2 * half) * M + l16;
#pragma unroll
    for (int m = 0; m < MT; ++m) {
      v2f b;
      b.x = wk[m * 16];
      b.y = wk[M + m * 16];
      acc[m] = __builtin_amdgcn_wmma_f32_16x16x4_f32(
          false, a, false, b, (short)0, acc[m], false, false);
    }
  }

  int row = n0 + 8 * half;
#pragma unroll
  for (int m = 0; m < MT; ++m)
#pragma unroll
    for (int r = 0; r < 8; ++r)
      H[(size_t)(row + r) * M + m * 16 + l16] = acc[m][r];
}

// ---------------- self-loop init: o = h * dinv[i]^2 (float4 over features) --
__global__ __launch_bounds__(256) void k_selfscale(const float4* __restrict__ h,
                                                   const float* __restrict__ dinv,
                                                   float4* __restrict__ o,
                                                   int n, int f4) {
  int idx = blockIdx.x * 256 + threadIdx.x;
  if (idx >= n * f4) return;
  float s = dinv[idx / f4];
  s *= s;
  float4 v = h[idx];
  o[idx] = make_float4(v.x * s, v.y * s, v.z * s, v.w * s);
}

// ---------------- edge scatter-add: acc[dst] += h[src] * dinv[s]*dinv[d] ----
// one wave per edge; F=128 -> float4 per lane (4 atomics), F=32 -> 1 per lane
template <int F>
__global__ __launch_bounds__(256) void k_scatter(const float* __restrict__ h,
                                                 const int* __restrict__ src,
                                                 const int* __restrict__ dst,
                                                 const float* __restrict__ dinv,
                                                 float* __restrict__ acc, int E) {
  int gid = blockIdx.x * 256 + threadIdx.x;
  int e = gid >> 5;
  if (e >= E) return;
  int lane = gid & 31;
  int s = src[e], d = dst[e];
  float nrm = dinv[s] * dinv[d];
  if (F == 128) {
    float4 v = ((const float4*)(h + (size_t)s * F))[lane];
    float* ap = acc + (size_t)d * F + lane * 4;
    unsafeAtomicAdd(ap + 0, v.x * nrm);
    unsafeAtomicAdd(ap + 1, v.y * nrm);
    unsafeAtomicAdd(ap + 2, v.z * nrm);
    unsafeAtomicAdd(ap + 3, v.w * nrm);
  } else {
    unsafeAtomicAdd(acc + (size_t)d * F + lane, h[(size_t)s * F + lane] * nrm);
  }
}

// ---------------- bias + relu (in place) ----------------
__global__ __launch_bounds__(256) void k_biasrelu(float* __restrict__ a,
                                                  const float* __restrict__ b,
                                                  int total, int fmask) {
  int idx = blockIdx.x * 256 + threadIdx.x;
  if (idx >= total) return;
  float v = a[idx] + b[idx & fmask];
  a[idx] = fmaxf(v, 0.0f);
}

// ---------------- fused bias + log_softmax, one wave per node (32 classes) --
__global__ __launch_bounds__(256) void k_logsoftmax(float* __restrict__ o,
                                                    const float* __restrict__ b2,
                                                    int n) {
  int gid = blockIdx.x * 256 + threadIdx.x;
  int i = gid >> 5, lane = gid & 31;
  if (i >= n) return;
  float v = o[(size_t)i * 32 + lane] + b2[lane];
  float m = v;
#pragma unroll
  for (int off = 16; off; off >>= 1) m = fmaxf(m, __shfl_xor(m, off, 32));
  float p = expf(v - m);
  float ssum = p;
#pragma unroll
  for (int off = 16; off; off >>= 1) ssum += __shfl_xor(ssum, off, 32);
  o[(size_t)i * 32 + lane] = (v - m) - logf(ssum);
}

// ---------------- driver ----------------
extern "C" void kernel_launch(void* const* d_in, const int* in_sizes, int n_in,
                              void* d_out, int out_size, void* d_ws, size_t ws_size,
                              hipStream_t stream) {
  const float* x  = (const float*)d_in[0];
  const int*   ei = (const int*)d_in[1];
  const float* W1 = (const float*)d_in[2];
  const float* b1 = (const float*)d_in[3];
  const float* W2 = (const float*)d_in[4];
  const float* b2 = (const float*)d_in[5];
  float* out = (float*)d_out;

  const int n = N_NODES;
  const int E = in_sizes[1] / 2;
  const int* src = ei;
  const int* dst = ei + E;

  char* ws = (char*)d_ws;
  float* dinv = (float*)ws;                                           // N floats
  float* h1   = (float*)(ws + (256 << 10));                           // N*HID
  float* agg1 = (float*)(ws + (256 << 10) + (size_t)n * HID * 4);     // N*HID
  float* h2   = h1;  // h1 dead after layer-1 scatter; reuse for h2 (N*CLS)

  // normalization vector
  k_zero<<<(n + 255) / 256, 256, 0, stream>>>(dinv, n);
  k_count<<<(E + 255) / 256, 256, 0, stream>>>(dst, dinv, E);
  k_rsqrt<<<(n + 255) / 256, 256, 0, stream>>>(dinv, n);

  const int nstrips = n / 16;  // 50000 = 3125 * 16
  const int gemm_blocks = (nstrips + 3) / 4;

  // layer 1
  k_gemm<F_IN, HID><<<gemm_blocks, 128, 0, stream>>>(x, W1, h1, nstrips);
  k_selfscale<<<((n * (HID / 4)) + 255) / 256, 256, 0, stream>>>(
      (const float4*)h1, dinv, (float4*)agg1, n, HID / 4);
  k_scatter<HID><<<((E * 32) + 255) / 256, 256, 0, stream>>>(h1, src, dst, dinv, agg1, E);
  k_biasrelu<<<((n * HID) + 255) / 256, 256, 0, stream>>>(agg1, b1, n * HID, HID - 1);

  // layer 2
  k_gemm<HID, CLS><<<gemm_blocks, 128, 0, stream>>>(agg1, W2, h2, nstrips);
  k_selfscale<<<((n * (CLS / 4)) + 255) / 256, 256, 0, stream>>>(
      (const float4*)h2, dinv, (float4*)out, n, CLS / 4);
  k_scatter<CLS><<<((E * 32) + 255) / 256, 256, 0, stream>>>(h2, src, dst, dinv, out, E);
  k_logsoftmax<<<((n * 32) + 255) / 256, 256, 0, stream>>>(out, b2, n);
}